// RotaryPositionalEmbeddings2D_32495722561963
// MI455X (gfx1250) — compile-verified
//
#include <hip/hip_runtime.h>
#include <math.h>

#define THREADS     256
#define TPB         8          // tiles per block (tile = one (b,t) = 16 heads * 64 = 1024 floats)
#define TILE_FLOATS 1024
#define HALF        32         // pairs per head

typedef float v4f __attribute__((ext_vector_type(4)));
typedef int   v4i __attribute__((__vector_size__(4 * sizeof(int))));

#if __has_builtin(__builtin_amdgcn_global_load_async_to_lds_b128) && \
    __has_builtin(__builtin_amdgcn_s_wait_asynccnt)
#define USE_ASYNC 1
#else
#define USE_ASYNC 0
#endif

#define AS_GLOBAL __attribute__((address_space(1)))
#define AS_LDS    __attribute__((address_space(3)))

__global__ __launch_bounds__(THREADS)
void rope2d_kernel(const float* __restrict__ x,
                   const int* __restrict__ sotp,
                   const int* __restrict__ eovp,
                   const int* __restrict__ imghp,
                   const int* __restrict__ imgwp,
                   float* __restrict__ out,
                   int totalTiles, int S)
{
    __shared__ __attribute__((aligned(16))) float xbuf[2][TILE_FLOATS];
    __shared__ __attribute__((aligned(16))) float csbuf[2][2 * HALF];

    const int tid     = threadIdx.x;
    const int img_w   = imgwp[0];
    const int img_len = imghp[0] * img_w;

    const int tile0  = blockIdx.x * TPB;
    const int ntiles = min(TPB, totalTiles - tile0);
    if (ntiles <= 0) return;

    // lanes < 32: compute the 32 (cos,sin) pairs for one tile into csbuf[buf]
    auto compute_cs = [&](int tile, int buf) {
        if (tid < HALF) {
            int b   = tile / S;
            int t   = tile - b * S;
            int sot = sotp[b];
            int eov = min(eovp[b], S);
            bool valid   = (sot >= 0) && (eov > sot);
            int img_end  = min(eov, sot + img_len);
            bool in_img  = valid && (t >= sot) && (t < img_end);
            int pos      = in_img ? (t - sot) / img_w : t;
            // inv[j] = 10000^(-2j/64) = 2^(-j * log2(10000)/32)
            float inv = exp2f(-(float)tid * 0.41524101186092437f);
            float ang = (float)pos * inv;
            float sn, cs;
            sincosf(ang, &sn, &cs);
            csbuf[buf][2 * tid]     = cs;
            csbuf[buf][2 * tid + 1] = sn;
        }
    };

    // ---- prologue: stage tile 0 ----
#if USE_ASYNC
    __builtin_amdgcn_global_load_async_to_lds_b128(
        (AS_GLOBAL v4i*)(x + (size_t)tile0 * TILE_FLOATS + tid * 4),
        (AS_LDS v4i*)&xbuf[0][tid * 4], 0, 0);
#endif
    compute_cs(tile0, 0);

    for (int i = 0; i < ntiles; ++i) {
        const int p = i & 1;
        const bool has_next = (i + 1) < ntiles;

#if USE_ASYNC
        if (has_next) {
            __builtin_amdgcn_global_load_async_to_lds_b128(
                (AS_GLOBAL v4i*)(x + (size_t)(tile0 + i + 1) * TILE_FLOATS + tid * 4),
                (AS_LDS v4i*)&xbuf[p ^ 1][tid * 4], 0, 0);
            __builtin_amdgcn_s_wait_asynccnt(1);   // tile i complete (in-order)
        } else {
            __builtin_amdgcn_s_wait_asynccnt(0);   // drain last tile
        }
#endif
        __syncthreads();   // publish csbuf[p] (written last iter / prologue)

        // cos/sin pairs for this thread's two complex pairs (16B aligned)
        v4f cs = *(const v4f*)&csbuf[p][(4 * tid) & 63];

#if USE_ASYNC
        v4f xv = *(const v4f*)&xbuf[p][tid * 4];                       // ds_load_b128
#else
        v4f xv = __builtin_nontemporal_load(
            (const v4f*)(x + (size_t)(tile0 + i) * TILE_FLOATS + tid * 4));
#endif
        v4f o;
        o.x = xv.x * cs.x - xv.y * cs.y;
        o.y = xv.x * cs.y + xv.y * cs.x;
        o.z = xv.z * cs.z - xv.w * cs.w;
        o.w = xv.z * cs.w + xv.w * cs.z;

        __builtin_nontemporal_store(
            o, (v4f*)(out + (size_t)(tile0 + i) * TILE_FLOATS + tid * 4));

        // prepare next tile's table in the other buffer; next iteration's
        // barrier orders these writes against next iteration's reads.
        if (has_next) compute_cs(tile0 + i + 1, p ^ 1);
    }
}

extern "C" void kernel_launch(void* const* d_in, const int* in_sizes, int n_in,
                              void* d_out, int out_size, void* d_ws, size_t ws_size,
                              hipStream_t stream) {
    const float* x   = (const float*)d_in[0];
    const int*   sot = (const int*)d_in[1];
    const int*   eov = (const int*)d_in[2];
    const int*   ih  = (const int*)d_in[3];
    const int*   iw  = (const int*)d_in[4];
    float*       out = (float*)d_out;

    const int n          = in_sizes[0];          // b*s*16*64
    const int B          = in_sizes[1];          // batch
    const int totalTiles = n / TILE_FLOATS;      // b*s
    const int S          = totalTiles / B;       // seq len

    const int grid = (totalTiles + TPB - 1) / TPB;
    rope2d_kernel<<<grid, THREADS, 0, stream>>>(x, sot, eov, ih, iw, out,
                                                totalTiles, S);
}